// MoERouter_20109036880141
// MI455X (gfx1250) — compile-verified
//
#include <hip/hip_runtime.h>
#include <hip/hip_bf16.h>
#include <math.h>

// Problem constants (match reference)
#define NTOK   16384
#define DMODEL 4096
#define NEXP   64

// Tiling
#define ROWS    32            // token rows per block (2 row tiles of 16)
#define BK      128           // K-chunk staged in LDS
#define NC      (DMODEL / BK) // 32 chunks
#define XSTRIDE (BK + 4)      // 132 floats: pad, keeps 16B row align for b128 async writes
#define WSTRIDE (NEXP + 4)    // 68 floats: pad, keeps 16B row align
#define LSTRIDE (NEXP + 4)

// LDS arena layout (float offsets); single shared object -> starts at LDS offset 0
#define XB0 0
#define XB1 (ROWS * XSTRIDE)
#define WB0 (2 * ROWS * XSTRIDE)
#define WB1 (WB0 + BK * WSTRIDE)
#define LG  (WB1 + BK * WSTRIDE)
#define SMEM_TOT (LG + ROWS * LSTRIDE)

// async-to-LDS ops issued per wave per chunk: 8 (x tile) + 16 (W tile)
#define APC (((ROWS * BK / 4) / 128) + ((BK * NEXP / 4) / 128))

typedef __attribute__((ext_vector_type(2))) float v2f;
typedef __attribute__((ext_vector_type(8))) float v8f;

// One wave32 instruction moves 32 lanes x 16B = 512B straight into LDS,
// tracked by ASYNCcnt (no VGPR round trip, no LOADcnt/DScnt pressure).
__device__ __forceinline__ void async_ld_b128(uint32_t lds_byte_off, const float* gaddr) {
    asm volatile("global_load_async_to_lds_b128 %0, %1, off"
                 :: "v"(lds_byte_off), "v"(gaddr) : "memory");
}
#define WAIT_ASYNC(n) asm volatile("s_wait_asynccnt %0" :: "i"(n) : "memory")

// Issue async staging of one K-chunk into the given ping-pong buffers.
__device__ __forceinline__ void stage_chunk(const float* __restrict__ x,
                                            const float* __restrict__ W,
                                            int rowBase, int k0,
                                            uint32_t xbase_bytes, uint32_t wbase_bytes,
                                            int tid) {
    // x tile: ROWS*BK floats = 1024 b128 granules, 8 per thread (coalesced)
    #pragma unroll
    for (int j = 0; j < (ROWS * BK / 4) / 128; ++j) {
        int lin = tid + j * 128;             // 0..1023
        int row = lin >> 5;                  // 32 granules per row
        int c4  = lin & 31;
        async_ld_b128(xbase_bytes + (uint32_t)(row * XSTRIDE + c4 * 4) * 4u,
                      x + (size_t)(rowBase + row) * DMODEL + k0 + c4 * 4);
    }
    // W tile: BK*64 floats = 2048 b128 granules, 16 per thread (coalesced)
    #pragma unroll
    for (int j = 0; j < (BK * NEXP / 4) / 128; ++j) {
        int lin = tid + j * 128;
        int row = lin >> 4;                  // 16 granules per row
        int c4  = lin & 15;
        async_ld_b128(wbase_bytes + (uint32_t)(row * WSTRIDE + c4 * 4) * 4u,
                      W + (size_t)(k0 + row) * NEXP + c4 * 4);
    }
}

__global__ __launch_bounds__(128) void moe_router_wmma(
    const float* __restrict__ x, const float* __restrict__ W,
    const float* __restrict__ bias, float* __restrict__ out)
{
    __shared__ __align__(16) float smem[SMEM_TOT];

    const int tid   = threadIdx.x;     // 0..127 (4 wave32)
    const int wave  = tid >> 5;        // expert tile
    const int lane  = tid & 31;
    const int half  = lane >> 4;       // WMMA 32-bit layout: lanes16-31 hold K+2,K+3
    const int r     = lane & 15;
    const int rowBase = blockIdx.x * ROWS;
    const int eBase   = wave * 16;

    v8f acc0 = {};   // rows 0..15
    v8f acc1 = {};   // rows 16..31

    // Prologue: prefetch chunk 0 into buffer 0
    stage_chunk(x, W, rowBase, 0, (uint32_t)XB0 * 4u, (uint32_t)WB0 * 4u, tid);

    for (int c = 0; c < NC; ++c) {
        const int cur = c & 1;
        // Overlap: issue chunk c+1 into the other buffer while we compute chunk c.
        if (c + 1 < NC) {
            stage_chunk(x, W, rowBase, (c + 1) * BK,
                        (cur ? (uint32_t)XB0 : (uint32_t)XB1) * 4u,
                        (cur ? (uint32_t)WB0 : (uint32_t)WB1) * 4u, tid);
            WAIT_ASYNC(APC);   // in-order completion: <=APC outstanding => chunk c landed
        } else {
            WAIT_ASYNC(0);
        }
        __syncthreads();       // all waves' async fills for chunk c visible

        const float* xs = smem + (cur ? XB1 : XB0);
        const float* ws = smem + (cur ? WB1 : WB0);

        #pragma unroll
        for (int kk = 0; kk < BK; kk += 4) {
            const int kc = kk + 2 * half;
            v2f a0 = *(const v2f*)(xs + r * XSTRIDE + kc);
            v2f a1 = *(const v2f*)(xs + (16 + r) * XSTRIDE + kc);
            v2f bb;
            bb.x = ws[(kc)     * WSTRIDE + eBase + r];
            bb.y = ws[(kc + 1) * WSTRIDE + eBase + r];
            acc0 = __builtin_amdgcn_wmma_f32_16x16x4_f32(
                false, a0, false, bb, (short)0, acc0, false, false);
            acc1 = __builtin_amdgcn_wmma_f32_16x16x4_f32(
                false, a1, false, bb, (short)0, acc1, false, false);
        }
        __syncthreads();       // protect ping-pong buffer before it is refilled
    }

    // ---- scatter logits (+bias) per C-matrix layout: VGPR i -> row i+8*half, col r
    const float bv = bias[eBase + r];
    float* lg = smem + LG;
    #pragma unroll
    for (int i = 0; i < 8; ++i) {
        lg[(i + 8 * half)      * LSTRIDE + eBase + r] = acc0[i] + bv;
        lg[(16 + i + 8 * half) * LSTRIDE + eBase + r] = acc1[i] + bv;
    }
    __syncthreads();

    // ---- per-row top-2 + renormalized softmax (partition function cancels)
    if (tid < ROWS) {
        const float* lrow = smem + LG + tid * LSTRIDE;
        float v1 = -INFINITY, v2 = -INFINITY;
        int   i1 = 0,         i2 = 0;
        #pragma unroll 8
        for (int e = 0; e < NEXP; ++e) {
            float v = lrow[e];
            if (v > v1)      { v2 = v1; i2 = i1; v1 = v; i1 = e; }  // strict > => lowest index wins ties (jax top_k)
            else if (v > v2) { v2 = v;  i2 = e; }
        }
        float e2 = expf(v2 - v1);          // <= 1, no overflow
        float p1 = 1.0f / (1.0f + e2);
        int row = rowBase + tid;
        out[row * 2 + 0] = p1;
        out[row * 2 + 1] = e2 * p1;
        float* oidx = out + (size_t)NTOK * 2;   // tuple output #2: indices
        oidx[row * 2 + 0] = (float)i1;
        oidx[row * 2 + 1] = (float)i2;
    }
}

extern "C" void kernel_launch(void* const* d_in, const int* in_sizes, int n_in,
                              void* d_out, int out_size, void* d_ws, size_t ws_size,
                              hipStream_t stream) {
    const float* x = (const float*)d_in[0];   // [16384, 4096]
    const float* W = (const float*)d_in[1];   // [4096, 64]
    const float* b = (const float*)d_in[2];   // [64]
    float* out = (float*)d_out;               // [16384,2] probs ++ [16384,2] indices
    (void)in_sizes; (void)n_in; (void)out_size; (void)d_ws; (void)ws_size;
    moe_router_wmma<<<NTOK / ROWS, 128, 0, stream>>>(x, W, b, out);
}